// MaskHead_50328426774933
// MI455X (gfx1250) — compile-verified
//
#include <hip/hip_runtime.h>
#include <hip/hip_bf16.h>

typedef __attribute__((ext_vector_type(2))) float v2f;
typedef __attribute__((ext_vector_type(4))) float v4f;
typedef __attribute__((ext_vector_type(8))) float v8f;

// ---------------- workspace layout (float offsets) ----------------
#define OFF_BG0 0        // 64*32
#define OFF_PT0 2048     // 64*32
#define OFF_BG1 4096     // 64*64
#define OFF_PT1 8192     // 64*64
#define OFF_BG2 12288    // 64*128
#define OFF_PT2 20480    // 64*128
#define OFF_V0  28672    // 32
#define OFF_V1  28704    // 64
#define OFF_V2  28768    // 128
#define OFF_BL  28896    // 3
#define OFF_S   28928    // 64
#define OFF_P0  28992    // 64
#define OFF_P1  29056    // 64
#define OFF_P2  29120    // 64
#define OFF_F0  29184    // 32768 (16B aligned: 29184 % 4 == 0)
#define OFF_F1  61952    // 4096
#define OFF_F2  66048    // 512
// total 66560 floats = 260 KB of d_ws

// ---------------- kernel A: v_lvl = w_log @ Wl_lvl, Bl_lvl = w_log . bl_lvl --
__global__ void precompute_v(const float* __restrict__ Wl0, const float* __restrict__ bl0,
                             const float* __restrict__ Wl1, const float* __restrict__ bl1,
                             const float* __restrict__ Wl2, const float* __restrict__ bl2,
                             const float* __restrict__ wlog, float* __restrict__ ws) {
    int t = threadIdx.x;
    for (int c = t; c < 32; c += 128) {
        float s = 0.f;
        for (int o = 0; o < 32; ++o) s = fmaf(wlog[o], Wl0[o * 32 + c], s);
        ws[OFF_V0 + c] = s;
    }
    for (int c = t; c < 64; c += 128) {
        float s = 0.f;
        for (int o = 0; o < 32; ++o) s = fmaf(wlog[o], Wl1[o * 64 + c], s);
        ws[OFF_V1 + c] = s;
    }
    for (int c = t; c < 128; c += 128) {
        float s = 0.f;
        for (int o = 0; o < 32; ++o) s = fmaf(wlog[o], Wl2[o * 128 + c], s);
        ws[OFF_V2 + c] = s;
    }
    if (t < 3) {
        const float* bl = (t == 0) ? bl0 : ((t == 1) ? bl1 : bl2);
        float s = 0.f;
        for (int o = 0; o < 32; ++o) s = fmaf(wlog[o], bl[o], s);
        ws[OFF_BL + t] = s;
    }
}

// ---------------- kernel B: six 64x192 @ 192xC GEMMs on WMMA f32 16x16x4 ----
struct GemmArgs {
    const float* W[6];     // Wb0,Wp0,Wb1,Wp1,Wb2,Wp2  (192 x C, row-major)
    const float* bias[6];  // bb0,bp0,bb1,bp1,bb2,bp2
    float*       out[6];   // BG0,PT0,BG1,PT1,BG2,PT2  (64 x C, row-major)
    int          C[6];
};

__global__ __launch_bounds__(32) void token_gemm_wmma(const float* __restrict__ tf,
                                                      GemmArgs ga) {
    // one wave == one 16x16 output tile; tiles per gemm = (64/16)*(C/16) = C/4
    int t = blockIdx.x;
    int g = 0, base = 0;
    while (g < 5) {
        int ntg = ga.C[g] >> 2;
        if (t < base + ntg) break;
        base += ntg;
        ++g;
    }
    int local = t - base;
    int C = ga.C[g];
    int sh = (C == 32) ? 1 : ((C == 64) ? 2 : 3);   // log2(C/16)
    int mt = local >> sh;
    int nt = local & ((1 << sh) - 1);

    const float* __restrict__ W = ga.W[g];
    int lane  = threadIdx.x;
    int l15   = lane & 15;
    int khalf = (lane >> 4) << 1;                   // lanes 16-31 handle K+2,K+3
    const float* __restrict__ arow = tf + ((mt << 4) + l15) * 192;
    const float* __restrict__ bcol = W + (nt << 4) + l15;

    v8f acc = {};
    for (int k0 = 0; k0 < 192; k0 += 4) {
        int ka = k0 + khalf;
        v2f a, b;
        a.x = arow[ka];
        a.y = arow[ka + 1];
        b.x = bcol[ka * C];
        b.y = bcol[(ka + 1) * C];
        acc = __builtin_amdgcn_wmma_f32_16x16x4_f32(false, a, false, b,
                                                    (short)0, acc, false, false);
    }

    float bias = ga.bias[g][(nt << 4) + l15];
    // C/D layout: VGPR r -> row r (lanes 0-15) / row r+8 (lanes 16-31), col = lane&15
    float* __restrict__ out =
        ga.out[g] + ((mt << 4) + ((lane >> 4) << 3)) * C + (nt << 4) + l15;
#pragma unroll
    for (int r = 0; r < 8; ++r) out[r * C] = acc[r] + bias;
}

// ---------------- kernel C: per-box scalars S[n], P_lvl[n] ------------------
__global__ void box_scalars(const float* __restrict__ blog, float* __restrict__ ws) {
    int n = threadIdx.x;  // 64 threads
    float a0 = 0.f, p0 = 0.f, a1 = 0.f, p1 = 0.f, a2 = 0.f, p2 = 0.f;
    for (int c = 0; c < 32; ++c) {
        float v = ws[OFF_V0 + c];
        a0 = fmaf(v, ws[OFF_BG0 + n * 32 + c], a0);
        p0 = fmaf(v, ws[OFF_PT0 + n * 32 + c], p0);
    }
    for (int c = 0; c < 64; ++c) {
        float v = ws[OFF_V1 + c];
        a1 = fmaf(v, ws[OFF_BG1 + n * 64 + c], a1);
        p1 = fmaf(v, ws[OFF_PT1 + n * 64 + c], p1);
    }
    for (int c = 0; c < 128; ++c) {
        float v = ws[OFF_V2 + c];
        a2 = fmaf(v, ws[OFF_BG2 + n * 128 + c], a2);
        p2 = fmaf(v, ws[OFF_PT2 + n * 128 + c], p2);
    }
    ws[OFF_S + n] = blog[0] + ws[OFF_BL] + ws[OFF_BL + 1] + ws[OFF_BL + 2] + a0 + a1 + a2;
    ws[OFF_P0 + n] = p0;
    ws[OFF_P1 + n] = p1;
    ws[OFF_P2 + n] = p2;
}

// ---------------- kernel D: F_lvl[p] = v_lvl . feat_lvl[:,p], 4 voxels/thread
__global__ __launch_bounds__(256) void fields(const float* __restrict__ f0,
                                              const float* __restrict__ f1,
                                              const float* __restrict__ f2,
                                              float* __restrict__ ws) {
    int i = blockIdx.x * 256 + threadIdx.x;   // vec4 index; 8192 + 1024 + 128 total
    if (i < 8192) {
        int p = i << 2;
        v4f s = {};
        for (int c = 0; c < 32; ++c) {
            v4f f = *(const v4f*)(f0 + c * 32768 + p);     // global_load_b128
            float v = ws[OFF_V0 + c];
            s.x = fmaf(v, f.x, s.x); s.y = fmaf(v, f.y, s.y);
            s.z = fmaf(v, f.z, s.z); s.w = fmaf(v, f.w, s.w);
        }
        *(v4f*)(ws + OFF_F0 + p) = s;
    } else if (i < 8192 + 1024) {
        int p = (i - 8192) << 2;
        v4f s = {};
        for (int c = 0; c < 64; ++c) {
            v4f f = *(const v4f*)(f1 + c * 4096 + p);
            float v = ws[OFF_V1 + c];
            s.x = fmaf(v, f.x, s.x); s.y = fmaf(v, f.y, s.y);
            s.z = fmaf(v, f.z, s.z); s.w = fmaf(v, f.w, s.w);
        }
        *(v4f*)(ws + OFF_F1 + p) = s;
    } else if (i < 8192 + 1024 + 128) {
        int p = (i - (8192 + 1024)) << 2;
        v4f s = {};
        for (int c = 0; c < 128; ++c) {
            v4f f = *(const v4f*)(f2 + c * 512 + p);
            float v = ws[OFF_V2 + c];
            s.x = fmaf(v, f.x, s.x); s.y = fmaf(v, f.y, s.y);
            s.z = fmaf(v, f.z, s.z); s.w = fmaf(v, f.w, s.w);
        }
        *(v4f*)(ws + OFF_F2 + p) = s;
    }
}

// ---------------- kernel E: fused upsample-sum + point deltas, 4 outs/thread -
__global__ __launch_bounds__(256) void final_sum(const float* __restrict__ ws,
                                                 const int* __restrict__ tokens,
                                                 float* __restrict__ out) {
    int i = blockIdx.x * 256 + threadIdx.x;   // 524288 vec4s = 2,097,152 outputs
    int base = i << 2;
    int n = base >> 15;
    int p = base & 32767;
    int d = p >> 10, h = (p >> 5) & 31, w = p & 31;   // w is 4-aligned

    // shared by all 4 outputs: same coarse (level-2) cell, same box
    float common = ws[OFF_S + n]
                 + ws[OFF_F2 + ((d >> 2) << 6) + ((h >> 2) << 3) + (w >> 2)];
    int f1idx = OFF_F1 + ((d >> 1) << 8) + ((h >> 1) << 4) + (w >> 1);
    float f1a = ws[f1idx];          // covers w, w+1
    float f1b = ws[f1idx + 1];      // covers w+2, w+3
    v4f f0 = *(const v4f*)(ws + OFF_F0 + p);   // global_load_b128

    v4f r;
    r.x = common + f1a + f0.x;
    r.y = common + f1a + f0.y;
    r.z = common + f1b + f0.z;
    r.w = common + f1b + f0.w;

    int x = tokens[n * 4 + 0], y = tokens[n * 4 + 1], z = tokens[n * 4 + 2];
    int mdh = (d ^ x) | (h ^ y);
    // (w+j)^z has identical bits >=2 for all j in 0..3 (w 4-aligned)
    if (!((mdh | (w ^ z)) >> 2)) {             // inside the 4x4x4 level-2 cell
        float P2 = ws[OFF_P2 + n], P1 = ws[OFF_P1 + n], P0 = ws[OFF_P0 + n];
#pragma unroll
        for (int j = 0; j < 4; ++j) {
            int m = mdh | ((w + j) ^ z);
            float add = P2;                    // level-2 applies to all 4 lanes here
            if (!(m >> 1)) {                   // inside the 2x2x2 level-1 cell
                add += P1;
                if (!m) add += P0;             // exact voxel
            }
            r[j] += add;
        }
    }
    *(v4f*)(out + base) = r;                   // global_store_b128
}

// ---------------- launch ----------------------------------------------------
extern "C" void kernel_launch(void* const* d_in, const int* in_sizes, int n_in,
                              void* d_out, int out_size, void* d_ws, size_t ws_size,
                              hipStream_t stream) {
    const float* feat0  = (const float*)d_in[0];
    const float* feat1  = (const float*)d_in[1];
    const float* feat2  = (const float*)d_in[2];
    const int*   tokens = (const int*)d_in[3];
    const float* tf     = (const float*)d_in[4];
    const float* Wp0 = (const float*)d_in[5],  *bp0 = (const float*)d_in[6];
    const float* Wb0 = (const float*)d_in[7],  *bb0 = (const float*)d_in[8];
    const float* Wl0 = (const float*)d_in[9],  *bl0 = (const float*)d_in[10];
    const float* Wp1 = (const float*)d_in[11], *bp1 = (const float*)d_in[12];
    const float* Wb1 = (const float*)d_in[13], *bb1 = (const float*)d_in[14];
    const float* Wl1 = (const float*)d_in[15], *bl1 = (const float*)d_in[16];
    const float* Wp2 = (const float*)d_in[17], *bp2 = (const float*)d_in[18];
    const float* Wb2 = (const float*)d_in[19], *bb2 = (const float*)d_in[20];
    const float* Wl2 = (const float*)d_in[21], *bl2 = (const float*)d_in[22];
    const float* wlog = (const float*)d_in[23];
    const float* blog = (const float*)d_in[24];
    float* ws  = (float*)d_ws;
    float* out = (float*)d_out;

    precompute_v<<<1, 128, 0, stream>>>(Wl0, bl0, Wl1, bl1, Wl2, bl2, wlog, ws);

    GemmArgs ga;
    ga.W[0] = Wb0; ga.bias[0] = bb0; ga.out[0] = ws + OFF_BG0; ga.C[0] = 32;
    ga.W[1] = Wp0; ga.bias[1] = bp0; ga.out[1] = ws + OFF_PT0; ga.C[1] = 32;
    ga.W[2] = Wb1; ga.bias[2] = bb1; ga.out[2] = ws + OFF_BG1; ga.C[2] = 64;
    ga.W[3] = Wp1; ga.bias[3] = bp1; ga.out[3] = ws + OFF_PT1; ga.C[3] = 64;
    ga.W[4] = Wb2; ga.bias[4] = bb2; ga.out[4] = ws + OFF_BG2; ga.C[4] = 128;
    ga.W[5] = Wp2; ga.bias[5] = bp2; ga.out[5] = ws + OFF_PT2; ga.C[5] = 128;
    token_gemm_wmma<<<112, 32, 0, stream>>>(tf, ga);   // 8+8+16+16+32+32 = 112 tiles

    box_scalars<<<1, 64, 0, stream>>>(blog, ws);
    fields<<<37, 256, 0, stream>>>(feat0, feat1, feat2, ws);  // 9344 vec4 tasks
    final_sum<<<2048, 256, 0, stream>>>(ws, tokens, out);     // 524288 vec4 outputs
}